// SinusoidEncoding_42898133353386
// MI455X (gfx1250) — compile-verified
//
#include <hip/hip_runtime.h>
#include <math.h>

// ---------------------------------------------------------------------------
// SinusoidEncoding: out[k, c] = x[k, c] + P[k, c]
//   P[k, 2i]   = sin(k * f_i),  P[k, 2i+1] = cos(k * f_i),
//   f_i = 10000^(-2i/d) = exp2(-(i/(d/2)) * log2(10000)),  d = 4096.
//
// Memory-bound: 256 MiB traffic -> ~11.5 us floor at 23.3 TB/s.
// CDNA5 path: per-wave TENSOR_LOAD_TO_LDS DMA of a 16x128-float subtile,
// overlapped with the sincos setup; consume via ds_load_b128; P generated by
// the angle-addition recurrence (4 FMAs per sin/cos pair per row); results
// stored directly with global_store_b128.
// ---------------------------------------------------------------------------

#define D_MODEL        4096
#define THREADS        256
#define ROWS_PER_BLOCK 16
#define COLS_PER_BLOCK (THREADS * 4)     // 1024 floats per block-row
#define WAVE_COLS      128               // 32 lanes * 4 floats
#define WAVE_TILE_F    (ROWS_PER_BLOCK * WAVE_COLS)   // 2048 floats = 8 KB

typedef unsigned int u32x4 __attribute__((ext_vector_type(4)));
typedef int          i32x4 __attribute__((ext_vector_type(4)));
typedef int          i32x8 __attribute__((ext_vector_type(8)));

#if defined(__HIP_DEVICE_COMPILE__) && __has_builtin(__builtin_amdgcn_tensor_load_to_lds)
#define SINENC_USE_TDM 1
#else
#define SINENC_USE_TDM 0
#endif

__global__ __launch_bounds__(THREADS)
void sinenc_stream_kernel(const float* __restrict__ x, float* __restrict__ out)
{
    const int tid     = threadIdx.x;
    const int colbase = blockIdx.x * COLS_PER_BLOCK + tid * 4;   // multiple of 4
    const int row0    = blockIdx.y * ROWS_PER_BLOCK;

#if SINENC_USE_TDM
    // ---- Issue the per-wave TDM load first so it overlaps the setup math ----
    __shared__ float tile[ROWS_PER_BLOCK * COLS_PER_BLOCK];      // 64 KB
    const int wave = tid >> 5;
    const int lane = tid & 31;

    float* wtile = &tile[wave * WAVE_TILE_F];
    const unsigned lds_off = (unsigned)(unsigned long long)wtile; // LDS byte offset

    const unsigned long long gaddr =
        (unsigned long long)x +
        ((unsigned long long)row0 * D_MODEL +
         (unsigned long long)blockIdx.x * COLS_PER_BLOCK +
         (unsigned long long)wave * WAVE_COLS) * sizeof(float);

#define RFL(v) __builtin_amdgcn_readfirstlane((int)(v))
    // D# group 0: count=1 | lds_addr | global_addr[56:0] | type=2
    u32x4 g0;
    g0[0] = (unsigned)RFL(1u);
    g0[1] = (unsigned)RFL(lds_off);
    g0[2] = (unsigned)RFL((unsigned)(gaddr & 0xFFFFFFFFull));
    g0[3] = (unsigned)RFL(((unsigned)(gaddr >> 32) & 0x01FFFFFFu) | 0x80000000u);

    // D# group 1: wg_mask=0, data_size=2 (4B), no barrier/iterate/pad,
    // tensor_dim0=WAVE_COLS, tensor_dim1=ROWS, tile_dim0=WAVE_COLS,
    // tile_dim1=ROWS, tile_dim2=0, dim0_stride=D_MODEL, dim1_stride=0.
    i32x8 g1;
    g1[0] = RFL(2u << 16);                 // data_size=2 at bits [17:16]
    g1[1] = RFL((unsigned)WAVE_COLS << 16);      // tensor_dim0[15:0] at bits [63:48]
    g1[2] = RFL((unsigned)ROWS_PER_BLOCK << 16); // tensor_dim1[15:0] at bits [95:80]
    g1[3] = RFL((unsigned)WAVE_COLS << 16);      // tile_dim0 at bits [127:112]
    g1[4] = RFL((unsigned)ROWS_PER_BLOCK);       // tile_dim1 at bits [143:128]
    g1[5] = RFL((unsigned)D_MODEL);              // tensor_dim0_stride[31:0]
    g1[6] = RFL(0);
    g1[7] = RFL(0);

    i32x4 gz = {0, 0, 0, 0};
#if __has_include(<hip/amd_detail/amd_gfx1250_TDM.h>)
    i32x8 gz8 = {0, 0, 0, 0, 0, 0, 0, 0};
    __builtin_amdgcn_tensor_load_to_lds(g0, g1, gz, gz, gz8, 0);
#else
    __builtin_amdgcn_tensor_load_to_lds(g0, g1, gz, gz, 0);
#endif
#undef RFL
#endif  // SINENC_USE_TDM

    // ---- Transcendental setup (executes under the DMA latency) ----
    const float i0 = (float)(colbase >> 1);
    const float i1 = i0 + 1.0f;
    const float L2_10000 = 13.287712379549449f;          // log2(10000)
    const float f0 = exp2f(-(i0 * (1.0f / 2048.0f)) * L2_10000);
    const float f1 = exp2f(-(i1 * (1.0f / 2048.0f)) * L2_10000);

    float s0, c0, s1, c1, sf0, cf0, sf1, cf1;
    sincosf((float)row0 * f0, &s0, &c0);
    sincosf((float)row0 * f1, &s1, &c1);
    sincosf(f0, &sf0, &cf0);
    sincosf(f1, &sf1, &cf1);

    float* __restrict__ op = out + (size_t)row0 * D_MODEL + colbase;

#if SINENC_USE_TDM
    // ---- Wait for this wave's tile, then stream LDS -> VGPR -> global ----
    __builtin_amdgcn_s_wait_tensorcnt(0);
    asm volatile("" ::: "memory");   // LDS now holds the tile; no reordering

    const float* lp = wtile + lane * 4;
#pragma unroll 4
    for (int r = 0; r < ROWS_PER_BLOCK; ++r) {
        float4 v = *(const float4*)lp;          // ds_load_b128 (conflict-free)
        v.x += s0;  v.y += c0;
        v.z += s1;  v.w += c1;
        *(float4*)op = v;                       // global_store_b128

        const float ns0 = fmaf(s0, cf0,  c0 * sf0);
        const float nc0 = fmaf(c0, cf0, -s0 * sf0);
        const float ns1 = fmaf(s1, cf1,  c1 * sf1);
        const float nc1 = fmaf(c1, cf1, -s1 * sf1);
        s0 = ns0; c0 = nc0; s1 = ns1; c1 = nc1;

        lp += WAVE_COLS;
        op += D_MODEL;
    }
#else
    // ---- Fallback: direct global streaming ----
    const float* __restrict__ xp = x + (size_t)row0 * D_MODEL + colbase;
#pragma unroll 4
    for (int r = 0; r < ROWS_PER_BLOCK; ++r) {
        __builtin_prefetch(xp + 2 * D_MODEL, 0, 1);
        float4 v = *(const float4*)xp;
        v.x += s0;  v.y += c0;
        v.z += s1;  v.w += c1;
        *(float4*)op = v;

        const float ns0 = fmaf(s0, cf0,  c0 * sf0);
        const float nc0 = fmaf(c0, cf0, -s0 * sf0);
        const float ns1 = fmaf(s1, cf1,  c1 * sf1);
        const float nc1 = fmaf(c1, cf1, -s1 * sf1);
        s0 = ns0; c0 = nc0; s1 = ns1; c1 = nc1;

        xp += D_MODEL;
        op += D_MODEL;
    }
#endif
}

// ---------------------------------------------------------------------------
extern "C" void kernel_launch(void* const* d_in, const int* in_sizes, int n_in,
                              void* d_out, int out_size, void* d_ws, size_t ws_size,
                              hipStream_t stream)
{
    (void)n_in; (void)out_size; (void)d_ws; (void)ws_size;

    const float* x   = (const float*)d_in[0];
    float*       out = (float*)d_out;

    const int total = in_sizes[0];            // 8192 * 4096
    const int rows  = total / D_MODEL;        // 8192

    dim3 grid(D_MODEL / COLS_PER_BLOCK,       // 4
              rows / ROWS_PER_BLOCK);         // 512
    dim3 block(THREADS);

    sinenc_stream_kernel<<<grid, block, 0, stream>>>(x, out);
}